// GCNIIBlock_1365799600618
// MI455X (gfx1250) — compile-verified
//
#include <hip/hip_runtime.h>
#include <hip/hip_bf16.h>

// ---------------------------------------------------------------------------
// GCNII block on MI455X (gfx1250, wave32):
//   tokens = transpose(x)  [B,N,C]
//   S = tokens @ tokens^T  -> fused per-row top-9 (kNN)   [WMMA f32 16x16x4,
//       B tiles staged to LDS by the Tensor Data Mover, double-buffered]
//   h = 0.9*mean(neigh) + 0.1*tokens0
//   out = 0.5*h + 0.5*(h @ W^T + b)                       [WMMA f32 16x16x4]
//   BN (batch stats) + residual + ReLU
// ---------------------------------------------------------------------------

typedef float v2f __attribute__((ext_vector_type(2)));
typedef float v8f __attribute__((ext_vector_type(8)));
typedef unsigned int v4u __attribute__((ext_vector_type(4)));
typedef int v8i __attribute__((ext_vector_type(8)));
typedef int v4i __attribute__((ext_vector_type(4)));

constexpr int Bn = 16;
constexpr int Cc = 128;
constexpr int Nn = 48 * 48;        // 2304
constexpr int Kk = 9;
constexpr int TOT = Bn * Nn;       // 36864 tokens total

#if __has_builtin(__builtin_amdgcn_tensor_load_to_lds) && __has_builtin(__builtin_amdgcn_s_wait_tensorcnt)
#define HAVE_TDM 1
#else
#define HAVE_TDM 0
#endif

// ------------------------- layout transpose [B,C,N] -> [B,N,C] --------------
__global__ __launch_bounds__(256) void transpose_kernel(const float* __restrict__ src,
                                                        float* __restrict__ dst) {
  __shared__ float T[32][33];
  const int b  = blockIdx.z;
  const int n0 = blockIdx.x * 32;
  const int c0 = blockIdx.y * 32;
  const int tx = threadIdx.x & 31;
  const int ty = threadIdx.x >> 5;    // 0..7
#pragma unroll
  for (int it = 0; it < 4; ++it) {
    const int c = c0 + ty + 8 * it;
    T[ty + 8 * it][tx] = src[((size_t)(b * Cc + c)) * Nn + n0 + tx];  // T[c_loc][n_loc]
  }
  __syncthreads();
#pragma unroll
  for (int it = 0; it < 4; ++it) {
    const int n = n0 + ty + 8 * it;
    dst[((size_t)(b * Nn + n)) * Cc + c0 + tx] = T[tx][ty + 8 * it];
  }
}

// ------------------------- squared norms per token ---------------------------
__global__ __launch_bounds__(256) void sqnorm_kernel(const float* __restrict__ tokT,
                                                     float* __restrict__ sq) {
  const int g = blockIdx.x * 256 + threadIdx.x;   // 0..36863
  const float4* p = (const float4*)(tokT + (size_t)g * Cc);
  float s = 0.f;
#pragma unroll
  for (int i = 0; i < 32; ++i) {
    float4 v = p[i];
    s += v.x * v.x + v.y * v.y + v.z * v.z + v.w * v.w;
  }
  sq[g] = s;
}

// ------------------------- TDM tile loader -----------------------------------
// Loads a 16-row x 128-float tile (rows contiguous, stride 128 floats) from
// global into LDS, inserting 4 DWORDs of padding after every 128 DWORDs so the
// LDS image has a 132-float row stride (bank-conflict-free fragment reads).
#if HAVE_TDM
__device__ __forceinline__ void tdm_load_tile(const float* gsrc, unsigned lds_byte_addr) {
  const unsigned long long ga = (unsigned long long)(uintptr_t)gsrc;
  v4u g0;
  g0.x = 1u;                                            // count=1, user mode
  g0.y = lds_byte_addr;                                 // lds_addr
  g0.z = (unsigned)(ga & 0xffffffffu);                  // global_addr[31:0]
  g0.w = (unsigned)((ga >> 32) & 0x01ffffffu) | (2u << 30); // addr[56:32] | type=2
  v8i g1;
  g1[0] = (int)((2u << 16)          // data_size = 4 bytes
              | (1u << 20)          // pad_enable
              | (6u << 22)          // pad_interval: 128 DWORDs
              | (3u << 25));        // pad_amount: 4 DWORDs
  g1[1] = (int)(128u << 16);        // tensor_dim0[15:0] = 128
  g1[2] = (int)(16u << 16);         // tensor_dim0 hi=0 | tensor_dim1[15:0] = 16
  g1[3] = (int)(128u << 16);        // tensor_dim1 hi=0 | tile_dim0 = 128
  g1[4] = 16;                       // tile_dim1 = 16, tile_dim2 = 0
  g1[5] = 128;                      // tensor_dim0_stride = 128 elements
  g1[6] = 0;
  g1[7] = 0;
  const v4i z4 = {0, 0, 0, 0};
  const v8i z8 = {0, 0, 0, 0, 0, 0, 0, 0};
  __builtin_amdgcn_tensor_load_to_lds(g0, g1, z4, z4, z8, 0);
}
#endif

// ------------------------- fused Gram + top-9 kNN ---------------------------
// grid: (N/128, B); block 256 = 8 waves; wave w owns rows [blk*128 + 16w, +16)
// rank(col) = sq[col] - 2*S[row,col]   (row norm constant per row)
__global__ __launch_bounds__(256) void knn_kernel(const float* __restrict__ tokT,
                                                  const float* __restrict__ sq,
                                                  int* __restrict__ idxb) {
  const int b    = blockIdx.y;
  const int tid  = threadIdx.x;
  const int wave = tid >> 5;
  const int lane = tid & 31;
  const int lmod = lane & 15;
  const int lhi  = lane >> 4;                    // 0 or 1
  const int row0 = blockIdx.x * 128 + wave * 16; // wave's row tile base

  const float* Tb  = tokT + (size_t)b * Nn * Cc;
  const float* sqb = sq + (size_t)b * Nn;

  __shared__ float Bt[2][16][132];               // double-buffered B tile (padded)
  __shared__ float scr[8][16][17];               // per-wave rank-tile transpose

  // Preload A panel (16 rows x 128 K) as WMMA fragments: 32 x v2f regs.
  v2f areg[32];
  {
    const float* arow = Tb + (size_t)(row0 + lmod) * Cc + 2 * lhi;
#pragma unroll
    for (int kk = 0; kk < 32; ++kk) areg[kk] = *(const v2f*)(arow + 4 * kk);
  }

  float bestd[Kk];
  int   besti[Kk];
#pragma unroll
  for (int j = 0; j < Kk; ++j) { bestd[j] = 3.0e38f; besti[j] = 0; }

#if HAVE_TDM
  const unsigned ldsB0 = (unsigned)(uintptr_t)&Bt[0][0][0];
  const unsigned ldsB1 = (unsigned)(uintptr_t)&Bt[1][0][0];
  if (wave == 0) tdm_load_tile(Tb, ldsB0);       // prologue: tile 0 -> buf 0
#endif

  for (int t = 0; t < Nn / 16; ++t) {            // 144 column tiles
    const int col0 = t * 16;
#if HAVE_TDM
    const int cur = t & 1;
    if (wave == 0) __builtin_amdgcn_s_wait_tensorcnt(0);  // tile t landed
    __syncthreads();                             // buf[cur] visible to all waves
    if (wave == 0 && t + 1 < Nn / 16)            // prefetch tile t+1 via TDM
      tdm_load_tile(Tb + (size_t)(col0 + 16) * Cc, cur ? ldsB0 : ldsB1);
#else
    const int cur = 0;
    __syncthreads();                             // previous tile fully consumed
    {                                            // cooperative staging (fallback)
#pragma unroll
      for (int j = 0; j < 2; ++j) {
        const int idx = tid + 256 * j;           // 512 float4 chunks
        const int row = idx >> 5;
        const int c4  = idx & 31;
        float4 v = *(const float4*)(Tb + (size_t)(col0 + row) * Cc + c4 * 4);
        *(float4*)&Bt[cur][row][c4 * 4] = v;
      }
    }
    __syncthreads();
#endif
    v8f acc = {0.f, 0.f, 0.f, 0.f, 0.f, 0.f, 0.f, 0.f};
#pragma unroll
    for (int kk = 0; kk < 32; ++kk) {
      v2f bfrag = *(const v2f*)&Bt[cur][lmod][4 * kk + 2 * lhi];
      acc = __builtin_amdgcn_wmma_f32_16x16x4_f32(false, areg[kk],
                                                  false, bfrag,
                                                  (short)0, acc,
                                                  false, false);
    }
    const float sqc = sqb[col0 + lmod];
    // D layout: VGPR v, lane -> (M = v + 8*lhi, N = lmod). Store rank tile.
#pragma unroll
    for (int v = 0; v < 8; ++v)
      scr[wave][v + 8 * lhi][lmod] = sqc - 2.0f * acc[v];
    __syncthreads();
    // lane handles row lmod, candidate cols col0 + 8*lhi + j
#pragma unroll
    for (int j = 0; j < 8; ++j) {
      const float d  = scr[wave][lmod][8 * lhi + j];
      const int   ci = col0 + 8 * lhi + j;
      float mx = bestd[0]; int mp = 0;
#pragma unroll
      for (int q = 1; q < Kk; ++q)
        if (bestd[q] > mx) { mx = bestd[q]; mp = q; }
      if (d < mx) {
#pragma unroll
        for (int q = 0; q < Kk; ++q)
          if (q == mp) { bestd[q] = d; besti[q] = ci; }
      }
    }
    __syncthreads();
  }

  // merge the two half-lists (lane l and l+16 hold disjoint column halves of row lmod)
  float od[Kk]; int oi[Kk];
#pragma unroll
  for (int j = 0; j < Kk; ++j) {
    od[j] = __shfl_xor(bestd[j], 16, 32);
    oi[j] = __shfl_xor(besti[j], 16, 32);
  }
  if (lane < 16) {
    float cd[2 * Kk]; int ci2[2 * Kk];
#pragma unroll
    for (int j = 0; j < Kk; ++j) {
      cd[j] = bestd[j];      ci2[j] = besti[j];
      cd[Kk + j] = od[j];    ci2[Kk + j] = oi[j];
    }
    const int row = row0 + lane;
    int* op = idxb + ((size_t)b * Nn + row) * Kk;
#pragma unroll
    for (int s = 0; s < Kk; ++s) {
      float mn = cd[0]; int mi = ci2[0]; int mp = 0;
#pragma unroll
      for (int q = 1; q < 2 * Kk; ++q)
        if (cd[q] < mn) { mn = cd[q]; mi = ci2[q]; mp = q; }
      op[s] = mi;
#pragma unroll
      for (int q = 0; q < 2 * Kk; ++q)
        if (q == mp) cd[q] = 3.0e38f;
    }
  }
}

// ------------------------- gather neighbors + h ------------------------------
// block: 256 threads = 2 tokens x 128 channels
__global__ __launch_bounds__(256) void gather_h_kernel(const float* __restrict__ tokT,
                                                       const float* __restrict__ tok0T,
                                                       const int* __restrict__ idxb,
                                                       float* __restrict__ hbuf) {
  const int g = blockIdx.x * 2 + (threadIdx.x >> 7);   // global token id
  const int c = threadIdx.x & 127;
  const int b = g / Nn;
  const int* ip = idxb + (size_t)g * Kk;
  const float* Tb = tokT + (size_t)b * Nn * Cc;
  float s = 0.f;
#pragma unroll
  for (int k = 0; k < Kk; ++k) s += Tb[(size_t)ip[k] * Cc + c];
  hbuf[(size_t)g * Cc + c] = 0.9f * (s * (1.0f / 9.0f)) + 0.1f * tok0T[(size_t)g * Cc + c];
}

// ------------------------- linear: out = 0.5h + 0.5(hW^T + b) ---------------
// grid 288; block 256 = 8 waves; wave owns 16 token rows, loops 8 d-tiles.
__global__ __launch_bounds__(256) void linear_kernel(const float* __restrict__ hbuf,
                                                     const float* __restrict__ Wl,
                                                     const float* __restrict__ bl,
                                                     float* __restrict__ outp) {
  const int tid  = threadIdx.x;
  const int wave = tid >> 5;
  const int lane = tid & 31;
  const int lmod = lane & 15;
  const int lhi  = lane >> 4;
  const int m0   = blockIdx.x * 128 + wave * 16;

  v2f areg[32];
  {
    const float* arow = hbuf + (size_t)(m0 + lmod) * Cc + 2 * lhi;
#pragma unroll
    for (int kk = 0; kk < 32; ++kk) areg[kk] = *(const v2f*)(arow + 4 * kk);
  }

  for (int dt = 0; dt < 8; ++dt) {
    const int d0 = dt * 16;
    const float* brow = Wl + (size_t)(d0 + lmod) * Cc + 2 * lhi;  // W[d, c] row-major
    v8f acc = {0.f, 0.f, 0.f, 0.f, 0.f, 0.f, 0.f, 0.f};
#pragma unroll
    for (int kk = 0; kk < 32; ++kk) {
      v2f bfrag = *(const v2f*)(brow + 4 * kk);
      acc = __builtin_amdgcn_wmma_f32_16x16x4_f32(false, areg[kk],
                                                  false, bfrag,
                                                  (short)0, acc,
                                                  false, false);
    }
    const int d = d0 + lmod;
    const float bb = bl[d];
#pragma unroll
    for (int v = 0; v < 8; ++v) {
      const int m = m0 + v + 8 * lhi;
      const float hv = hbuf[(size_t)m * Cc + d];
      outp[(size_t)m * Cc + d] = 0.5f * hv + 0.5f * (acc[v] + bb);
    }
  }
}

// ------------------------- BN stats ------------------------------------------
__global__ void init_stats_kernel(float* stats) { stats[threadIdx.x] = 0.f; }

__global__ __launch_bounds__(256) void bn_stats_kernel(const float* __restrict__ outp,
                                                       float* __restrict__ stats) {
  const int c    = threadIdx.x & 127;
  const int half = threadIdx.x >> 7;
  const int base = blockIdx.x * 288;           // 128 blocks * 288 rows = 36864
  float s = 0.f, s2 = 0.f;
  for (int i = half; i < 288; i += 2) {
    const float v = outp[(size_t)(base + i) * Cc + c];
    s += v; s2 += v * v;
  }
  __shared__ float sh[2][128], sh2[2][128];
  sh[half][c] = s; sh2[half][c] = s2;
  __syncthreads();
  if (half == 0) {
    atomicAdd(&stats[c],        s  + sh[1][c]);
    atomicAdd(&stats[128 + c],  s2 + sh2[1][c]);
  }
}

__global__ void bn_final_kernel(const float* __restrict__ stats,
                                const float* __restrict__ gamma,
                                const float* __restrict__ beta,
                                float* __restrict__ ss) {
  const int c = threadIdx.x;                   // 128
  const float inv  = 1.0f / (float)TOT;
  const float mean = stats[c] * inv;
  const float var  = stats[128 + c] * inv - mean * mean;
  const float rstd = rsqrtf(var + 1e-5f);
  const float sc   = gamma[c] * rstd;
  ss[c]       = sc;
  ss[128 + c] = beta[c] - mean * sc;
}

// ------------------------- apply BN + residual + ReLU ------------------------
__global__ __launch_bounds__(256) void apply_kernel(const float* __restrict__ outp,
                                                    const float* __restrict__ ss,
                                                    const float* __restrict__ x,
                                                    float* __restrict__ out) {
  __shared__ float T[32][33];
  const int b  = blockIdx.z;
  const int n0 = blockIdx.x * 32;
  const int c0 = blockIdx.y * 32;
  const int tx = threadIdx.x & 31;
  const int ty = threadIdx.x >> 5;
#pragma unroll
  for (int it = 0; it < 4; ++it) {
    const int n = n0 + ty + 8 * it;
    T[ty + 8 * it][tx] = outp[((size_t)(b * Nn + n)) * Cc + c0 + tx];  // T[n_loc][c_loc]
  }
  __syncthreads();
#pragma unroll
  for (int it = 0; it < 4; ++it) {
    const int c = c0 + ty + 8 * it;
    const int n = n0 + tx;
    const float sc = ss[c], shf = ss[128 + c];
    const size_t xi = ((size_t)(b * Cc + c)) * Nn + n;
    const float v = sc * T[tx][ty + 8 * it] + shf + x[xi];
    out[xi] = v > 0.f ? v : 0.f;
  }
}

// ---------------------------------------------------------------------------
extern "C" void kernel_launch(void* const* d_in, const int* in_sizes, int n_in,
                              void* d_out, int out_size, void* d_ws, size_t ws_size,
                              hipStream_t stream) {
  const float* x     = (const float*)d_in[0];
  const float* x0    = (const float*)d_in[1];
  const float* Wl    = (const float*)d_in[2];
  const float* bl    = (const float*)d_in[3];
  const float* gamma = (const float*)d_in[4];
  const float* beta  = (const float*)d_in[5];
  float* out = (float*)d_out;

  // workspace carve-up
  float* tokT  = (float*)d_ws;                    // [B,N,C]
  float* tok0T = tokT  + (size_t)TOT * Cc;        // [B,N,C]
  float* hbuf  = tok0T + (size_t)TOT * Cc;        // [B*N,C]
  float* outp  = hbuf  + (size_t)TOT * Cc;        // [B*N,C] pre-BN
  float* sqv   = outp  + (size_t)TOT * Cc;        // [B*N]
  int*   idxb  = (int*)(sqv + TOT);               // [B*N,9]
  float* stats = (float*)(idxb + (size_t)TOT * Kk); // 256
  float* ss    = stats + 256;                     // 256 (scale, shift)

  dim3 tgrid(Nn / 32, Cc / 32, Bn);               // (72, 4, 16)
  transpose_kernel<<<tgrid, 256, 0, stream>>>(x, tokT);
  transpose_kernel<<<tgrid, 256, 0, stream>>>(x0, tok0T);

  sqnorm_kernel<<<TOT / 256, 256, 0, stream>>>(tokT, sqv);

  dim3 kgrid(Nn / 128, Bn);                       // (18, 16)
  knn_kernel<<<kgrid, 256, 0, stream>>>(tokT, sqv, idxb);

  gather_h_kernel<<<TOT / 2, 256, 0, stream>>>(tokT, tok0T, idxb, hbuf);

  linear_kernel<<<TOT / 128, 256, 0, stream>>>(hbuf, Wl, bl, outp);

  init_stats_kernel<<<1, 256, 0, stream>>>(stats);
  bn_stats_kernel<<<128, 256, 0, stream>>>(outp, stats);
  bn_final_kernel<<<1, 128, 0, stream>>>(stats, gamma, beta, ss);

  apply_kernel<<<tgrid, 256, 0, stream>>>(outp, ss, x, out);
}